// RoPEDemoAttention_21010980012255
// MI455X (gfx1250) — compile-verified
//
#include <hip/hip_runtime.h>
#include <math.h>

typedef __attribute__((ext_vector_type(2))) float v2f;
typedef __attribute__((ext_vector_type(8))) float v8f;

// D = A(16x4) * B(4x16) + C(16x16), fp32, wave32. Exact fp32 matrix math.
__device__ __forceinline__ v8f wmma_f32(v2f a, v2f b, v8f c) {
    return __builtin_amdgcn_wmma_f32_16x16x4_f32(
        /*neg_a=*/false, a, /*neg_b=*/false, b,
        /*c_mod=*/(short)0, c, /*reuse_a=*/false, /*reuse_b=*/false);
}

// ln(10000)/128 : inv_freq[i] = 10000^(-2i/256) = exp(-i * LN_BASE_O128)
#define LN_BASE_O128 0.07195578415785784f

// ---------------------------------------------------------------------------
// Unified NT GEMM: C = X * Y^T (contract over last dim of both; both
// row-major with leading dim == K, so every A/B fragment is one coalesced
// 8-byte load per lane, matching the 32-bit WMMA A/B VGPR layout:
//   lanes 0-15 -> K = {0,1}, lanes 16-31 -> K = {2,3}).
//
// Wave tile 32(M) x 64(N): 8 f32 accumulators, 2 A-frags, 4 B-frags per
// K-step (8 WMMAs per 6 b64 loads). K-loop is software-pipelined with a
// register double buffer so next-step loads issue before current WMMAs.
//
// ROPE:   fuse rotary embedding into the epilogue (partner column n^1 lives
//         in the adjacent lane -> __shfl_xor(.,1,32), wave32).
// TSTORE: store C transposed per batch (C[b, n, s]) — materializes V^T so
//         the output GEMM is also a fast NT GEMM.
//
// Block: 256 thr = 8 waves. Block tile 256(M) x 64(N).
// Grid: (N/64, M/256, batches). All dims divide tiles exactly -> no
// predication, EXEC all-ones around WMMA (ISA requirement).
// ---------------------------------------------------------------------------
template <int ROPE, int TSTORE>
__global__ __launch_bounds__(256) void gemm_nt_wmma(
    const float* __restrict__ X, const float* __restrict__ Y,
    float* __restrict__ C, int M, int N, int K,
    long xStride, long yStride, long cStride, float scale, int seqS) {
    const int lane = threadIdx.x & 31;
    const int wave = threadIdx.x >> 5;   // 0..7
    const int half = lane >> 4;          // 0/1 -> K offset 0 / 2
    const int l    = lane & 15;

    const int m0 = blockIdx.y * 256 + wave * 32;
    const int n0 = blockIdx.x * 64;

    const float* Xb = X + (size_t)blockIdx.z * xStride;
    const float* Yb = Y + (size_t)blockIdx.z * yStride;

    v8f zero = {};
    v8f acc[2][4];
#pragma unroll
    for (int i = 0; i < 2; ++i)
#pragma unroll
        for (int t = 0; t < 4; ++t) acc[i][t] = zero;

    // Per-lane fragment base pointers (contiguous float2 along K).
    const float* ap0 = Xb + (size_t)(m0 + l) * K + 2 * half;
    const float* ap1 = ap0 + (size_t)16 * K;
    const float* bp0 = Yb + (size_t)(n0 +  0 + l) * K + 2 * half;
    const float* bp1 = Yb + (size_t)(n0 + 16 + l) * K + 2 * half;
    const float* bp2 = Yb + (size_t)(n0 + 32 + l) * K + 2 * half;
    const float* bp3 = Yb + (size_t)(n0 + 48 + l) * K + 2 * half;

    // ---- software pipeline: stage 0 prefetch ----
    v2f a0 = *(const v2f*)(ap0);
    v2f a1 = *(const v2f*)(ap1);
    v2f b0 = *(const v2f*)(bp0);
    v2f b1 = *(const v2f*)(bp1);
    v2f b2 = *(const v2f*)(bp2);
    v2f b3 = *(const v2f*)(bp3);

#pragma unroll 2
    for (int k0 = 0; k0 < K - 4; k0 += 4) {
        // prefetch next K-step while current WMMAs execute
        v2f na0 = *(const v2f*)(ap0 + k0 + 4);
        v2f na1 = *(const v2f*)(ap1 + k0 + 4);
        v2f nb0 = *(const v2f*)(bp0 + k0 + 4);
        v2f nb1 = *(const v2f*)(bp1 + k0 + 4);
        v2f nb2 = *(const v2f*)(bp2 + k0 + 4);
        v2f nb3 = *(const v2f*)(bp3 + k0 + 4);

        acc[0][0] = wmma_f32(a0, b0, acc[0][0]);
        acc[0][1] = wmma_f32(a0, b1, acc[0][1]);
        acc[0][2] = wmma_f32(a0, b2, acc[0][2]);
        acc[0][3] = wmma_f32(a0, b3, acc[0][3]);
        acc[1][0] = wmma_f32(a1, b0, acc[1][0]);
        acc[1][1] = wmma_f32(a1, b1, acc[1][1]);
        acc[1][2] = wmma_f32(a1, b2, acc[1][2]);
        acc[1][3] = wmma_f32(a1, b3, acc[1][3]);

        a0 = na0; a1 = na1;
        b0 = nb0; b1 = nb1; b2 = nb2; b3 = nb3;
    }
    // pipeline drain: last K-step
    acc[0][0] = wmma_f32(a0, b0, acc[0][0]);
    acc[0][1] = wmma_f32(a0, b1, acc[0][1]);
    acc[0][2] = wmma_f32(a0, b2, acc[0][2]);
    acc[0][3] = wmma_f32(a0, b3, acc[0][3]);
    acc[1][0] = wmma_f32(a1, b0, acc[1][0]);
    acc[1][1] = wmma_f32(a1, b1, acc[1][1]);
    acc[1][2] = wmma_f32(a1, b2, acc[1][2]);
    acc[1][3] = wmma_f32(a1, b3, acc[1][3]);

    // Output base: normal path is per-z-batch; transposed path derives the
    // batch from the flattened row index (the whole 32-row wave tile stays
    // in one batch since seqS is a multiple of the 256-row block tile).
    const int  bt   = TSTORE ? (m0 / seqS) : (int)blockIdx.z;
    const int  mrel = TSTORE ? (m0 - bt * seqS) : m0;
    float*     Cb   = C + (size_t)bt * cStride;

#pragma unroll
    for (int i = 0; i < 2; ++i) {
#pragma unroll
        for (int t = 0; t < 4; ++t) {
#pragma unroll
            for (int r = 0; r < 8; ++r) {
                const int m = m0 + 16 * i + r + 8 * half;  // global row
                const int n = n0 + t * 16 + l;             // global col
                float v = acc[i][t][r] * scale;
                if (ROPE) {
                    // partner lane holds column n^1 at the same row
                    float p   = __shfl_xor(v, 1, 32);
                    int   pos = m % seqS;
                    int   fi  = n & 127;    // emb = concat([freqs, freqs])
                    float ang =
                        (float)pos * __expf(-(float)fi * LN_BASE_O128);
                    float cs, sn;
                    __sincosf(ang, &sn, &cs);
                    // rotate_half: even n -> -x[n+1], odd n -> +x[n-1]
                    float rot = (n & 1) ? p : -p;
                    v = v * cs + rot * sn;
                }
                if (TSTORE) {
                    Cb[(size_t)n * seqS + (mrel + 16 * i + r + 8 * half)] = v;
                } else {
                    Cb[(size_t)m * N + n] = v;
                }
            }
        }
    }
}

// ---------------------------------------------------------------------------
// In-place row softmax; one 256-thread block per row of 4096 elements.
// ---------------------------------------------------------------------------
__global__ __launch_bounds__(256) void softmax_rows(float* __restrict__ Wt,
                                                    int n) {
    float* row = Wt + (size_t)blockIdx.x * n;
    const int tid = threadIdx.x;
    __shared__ float red[256];

    float vals[16];
    float mx = -3.402823466e38f;
#pragma unroll
    for (int i = 0; i < 16; ++i) {
        vals[i] = row[tid + i * 256];
        mx = fmaxf(mx, vals[i]);
    }
    red[tid] = mx;
    __syncthreads();
    for (int s = 128; s > 0; s >>= 1) {
        if (tid < s) red[tid] = fmaxf(red[tid], red[tid + s]);
        __syncthreads();
    }
    mx = red[0];
    __syncthreads();

    float sum = 0.f;
#pragma unroll
    for (int i = 0; i < 16; ++i) {
        vals[i] = __expf(vals[i] - mx);
        sum += vals[i];
    }
    red[tid] = sum;
    __syncthreads();
    for (int s = 128; s > 0; s >>= 1) {
        if (tid < s) red[tid] += red[tid + s];
        __syncthreads();
    }
    const float inv = 1.0f / red[0];
#pragma unroll
    for (int i = 0; i < 16; ++i) row[tid + i * 256] = vals[i] * inv;
}

extern "C" void kernel_launch(void* const* d_in, const int* in_sizes, int n_in,
                              void* d_out, int out_size, void* d_ws,
                              size_t ws_size, hipStream_t stream) {
    const float* x  = (const float*)d_in[0];
    const float* wq = (const float*)d_in[1];
    const float* wk = (const float*)d_in[2];
    const float* wv = (const float*)d_in[3];

    const int B = 4, S = 4096, D = 256;
    const size_t BSD = (size_t)B * S * D;   // 4,194,304
    const size_t SS  = (size_t)S * S;       // 16,777,216

    float* out     = (float*)d_out;         // [B,S,D]
    float* weights = out + BSD;             // [B,S,S]

    float* q  = (float*)d_ws;               // [B,S,D]  (RoPE'd)
    float* k  = q + BSD;                    // [B,S,D]  (RoPE'd)
    float* vt = k + BSD;                    // [B,D,S]  (V transposed)

    dim3 blk(256);
    const float inv_sqrt_d = 1.0f / 16.0f;  // 1/sqrt(256)

    // 1) QKV projections (M = B*S flat, NT vs weight matrices).
    //    q,k get fused RoPE; v is stored transposed per batch.
    dim3 gQKV(D / 64, (B * S) / 256, 1);
    gemm_nt_wmma<1, 0><<<gQKV, blk, 0, stream>>>(x, wq, q, B * S, D, D,
                                                 0L, 0L, 0L, 1.0f, S);
    gemm_nt_wmma<1, 0><<<gQKV, blk, 0, stream>>>(x, wk, k, B * S, D, D,
                                                 0L, 0L, 0L, 1.0f, S);
    gemm_nt_wmma<0, 1><<<gQKV, blk, 0, stream>>>(x, wv, vt, B * S, D, D,
                                                 0L, 0L, (long)D * S, 1.0f, S);

    // 2) scores = q @ k^T / sqrt(D), per batch, straight into the weights
    //    region of d_out (268 MB — the one mandatory HBM write; q/k/vt all
    //    stay resident in the 192 MB L2).
    dim3 gSc(S / 64, S / 256, B);
    gemm_nt_wmma<0, 0><<<gSc, blk, 0, stream>>>(q, k, weights, S, S, D,
                                                (long)S * D, (long)S * D,
                                                (long)SS, inv_sqrt_d, S);

    // 3) softmax over each of the B*S rows (in place in d_out).
    softmax_rows<<<B * S, blk, 0, stream>>>(weights, S);

    // 4) output = weights @ v  ==  weights (NT) vt, per batch: all fragment
    //    loads are coalesced b64, same fast path as the scores kernel.
    dim3 gOut(D / 64, S / 256, B);
    gemm_nt_wmma<0, 0><<<gOut, blk, 0, stream>>>(weights, vt, out, S, D, S,
                                                 (long)SS, (long)D * S,
                                                 (long)S * D, 1.0f, S);
}